// OctahedralCavityProcessor_73547019976727
// MI455X (gfx1250) — compile-verified
//
#include <hip/hip_runtime.h>
#include <hip/hip_bf16.h>

// ---------------------------------------------------------------------------
// Problem constants (match reference)
// ---------------------------------------------------------------------------
#define BB 8
#define CC 128
#define PP 100000
#define KK 14
#define HH 8
#define DH 16            // CC / HH
#define C2 256           // 2*CC
#define RAD2 0.25f       // RADIUS^2

typedef __attribute__((ext_vector_type(2))) float v2f;
typedef __attribute__((ext_vector_type(8))) float v8f;

// ---------------------------------------------------------------------------
// Kernel 1: per-point prep.  For each point: 14-bit in-radius mask word,
// nearest-cavity index (first-min semantics like jnp.argmin), and per-cavity
// point counts (LDS-reduced, then one global atomic per block).
// ---------------------------------------------------------------------------
__global__ __launch_bounds__(256)
void prep_kernel(const float* __restrict__ points, const float* __restrict__ cavities,
                 unsigned* __restrict__ maskbits, int* __restrict__ nearest,
                 float* __restrict__ counts) {
  __shared__ float cav[KK * 3];
  __shared__ float lcnt[16];
  int tid = threadIdx.x;
  if (tid < KK * 3) cav[tid] = cavities[tid];
  if (tid < 16) lcnt[tid] = 0.0f;
  __syncthreads();

  int p = blockIdx.x * 256 + tid;
  if (p < PP) {
    float px = points[p * 3 + 0];
    float py = points[p * 3 + 1];
    float pz = points[p * 3 + 2];
    unsigned bits = 0u;
    float best = 3.0e38f;
    int bi = 0;
#pragma unroll
    for (int k = 0; k < KK; ++k) {
      float dx = px - cav[k * 3 + 0];
      float dy = py - cav[k * 3 + 1];
      float dz = pz - cav[k * 3 + 2];
      float d2 = dx * dx + dy * dy + dz * dz;
      if (d2 < RAD2) {
        bits |= (1u << k);
        atomicAdd(&lcnt[k], 1.0f);
      }
      if (d2 < best) { best = d2; bi = k; }   // strict < keeps first minimum
    }
    maskbits[p] = bits;
    nearest[p] = bi;
  }
  __syncthreads();
  if (tid < KK) atomicAdd(&counts[tid], lcnt[tid]);
}

// ---------------------------------------------------------------------------
// Kernel 2: masked pooling GEMM via V_WMMA_F32_16X16X4_F32.
//   sums[b,k,c] = sum_p x[b,c,p] * mask[k,p]
// Block = 256 threads (8 waves). Each block owns one (b, 16-row C tile) and a
// strided subset of 256-wide P tiles. x tile staged in LDS (coalesced loads,
// padded stride 260 -> conflict-free column reads). Each wave holds one
// 16(C)x16(K) f32 accumulator across all its P tiles, then atomic-adds once.
// ---------------------------------------------------------------------------
#define PBLK 256
#define XSTR 260   // PBLK + 4 pad: row-to-row bank offset = 4 -> 16 distinct banks

__global__ __launch_bounds__(256)
void pool_kernel(const float* __restrict__ x, const unsigned* __restrict__ maskbits,
                 float* __restrict__ sums) {
  __shared__ float    xt[16 * XSTR];
  __shared__ unsigned mt[PBLK];

  const int ctiles = CC / 16;               // 8
  int bc   = blockIdx.x;                    // 0 .. BB*ctiles-1
  int b    = bc / ctiles;
  int c0   = (bc % ctiles) * 16;
  int tid  = threadIdx.x;
  int lane = tid & 31;
  int wave = tid >> 5;                      // 0..7
  int lo   = lane & 15;                     // A: C row  /  B: cavity col
  int hi   = lane >> 4;                     // K-phase selector (0 or 1)

  v8f acc = {};                             // 16x16 f32 accumulator (8 VGPRs)

  const int ntiles = (PP + PBLK - 1) / PBLK;
  const float* xb = x + (size_t)b * CC * PP;

  for (int t = blockIdx.y; t < ntiles; t += gridDim.y) {
    int pbase = t * PBLK;
    __syncthreads();                        // protect LDS reuse across iterations
    // stage x tile: 16 C-rows x PBLK points, coalesced per row
#pragma unroll
    for (int r = 0; r < 16; ++r) {
      int p = pbase + tid;
      xt[r * XSTR + tid] = (p < PP) ? xb[(size_t)(c0 + r) * PP + p] : 0.0f;
    }
    {
      int p = pbase + tid;
      mt[tid] = (p < PP) ? maskbits[p] : 0u;
    }
    __syncthreads();

    // 64 chunks of 4 points; wave w takes chunks w, w+8, ... (uniform per wave)
#pragma unroll
    for (int ch = 0; ch < 8; ++ch) {
      int p4 = (wave + ch * 8) * 4;
      // A fragment (16x4 f32): VGPR0 = K{0,2}, VGPR1 = K{1,3} per ISA layout
      v2f a;
      a.x = xt[lo * XSTR + p4 + 2 * hi + 0];
      a.y = xt[lo * XSTR + p4 + 2 * hi + 1];
      // B fragment (4x16 f32): lane = cavity col, same K striping as A
      unsigned m0 = mt[p4 + 2 * hi + 0];
      unsigned m1 = mt[p4 + 2 * hi + 1];
      v2f bfrag;
      bfrag.x = (float)((m0 >> lo) & 1u);
      bfrag.y = (float)((m1 >> lo) & 1u);
      acc = __builtin_amdgcn_wmma_f32_16x16x4_f32(
          /*neg_a=*/false, a, /*neg_b=*/false, bfrag,
          /*c_mod=*/(short)0, acc, /*reuse_a=*/false, /*reuse_b=*/false);
    }
  }

  // D layout: lane n = lo = cavity, VGPR j -> C row = 8*hi + j
  if (lo < KK) {
    float* dst = sums + ((size_t)b * KK + lo) * CC + c0 + 8 * hi;
#pragma unroll
    for (int j = 0; j < 8; ++j) atomicAdd(dst + j, acc[j]);
  }
}

// ---------------------------------------------------------------------------
// Kernel 3: feat = sums/counts, then per-cavity MLP: tanh(relu(f W1^T+b1) W2^T+b2)
// One block per (b,k).
// ---------------------------------------------------------------------------
__global__ __launch_bounds__(256)
void mlp_kernel(const float* __restrict__ sums, const float* __restrict__ counts,
                const float* __restrict__ w1, const float* __restrict__ b1,
                const float* __restrict__ w2, const float* __restrict__ b2,
                float* __restrict__ proc) {
  int bk = blockIdx.x;
  int b = bk / KK, k = bk % KK;
  __shared__ float feat[CC];
  __shared__ float hbuf[C2];
  int tid = threadIdx.x;

  float cnt = counts[k];
  float inv = (cnt > 0.0f) ? 1.0f / cnt : 0.0f;
  if (tid < CC) feat[tid] = sums[(size_t)bk * CC + tid] * inv;
  __syncthreads();

  {  // h[d] = relu(feat . w1[k,d,:] + b1[k,d]), d = tid in [0,256)
    const float* w = w1 + ((size_t)k * C2 + tid) * CC;
    float s = b1[k * C2 + tid];
    for (int c = 0; c < CC; ++c) s += feat[c] * w[c];
    hbuf[tid] = fmaxf(s, 0.0f);
  }
  __syncthreads();

  if (tid < CC) {  // proc[c] = tanh(h . w2[k,c,:] + b2[k,c])
    const float* w = w2 + ((size_t)k * CC + tid) * C2;
    float s = b2[k * CC + tid];
    for (int d = 0; d < C2; ++d) s += hbuf[d] * w[d];
    proc[(size_t)bk * CC + tid] = tanhf(s);
  }
}

// ---------------------------------------------------------------------------
// Kernel 4: multi-head self-attention over K=14 cavities + output projection.
// One block per batch b; everything lives in LDS (~41 KB of 320 KB).
// ---------------------------------------------------------------------------
__global__ __launch_bounds__(256)
void attn_kernel(const float* __restrict__ proc, const float* __restrict__ in_w,
                 const float* __restrict__ in_b, const float* __restrict__ out_w,
                 const float* __restrict__ out_b, float* __restrict__ attended) {
  int b = blockIdx.x;
  __shared__ float pr[KK * CC];        // 1792
  __shared__ float qkv[KK * 3 * CC];   // 5376
  __shared__ float sc[HH * KK * KK];   // 1568
  __shared__ float ob[KK * CC];        // 1792
  int tid = threadIdx.x;

  for (int i = tid; i < KK * CC; i += 256) pr[i] = proc[(size_t)b * KK * CC + i];
  __syncthreads();

  for (int i = tid; i < KK * 3 * CC; i += 256) {  // qkv = proc @ in_w^T + in_b
    int k = i / (3 * CC), e = i % (3 * CC);
    const float* w = in_w + (size_t)e * CC;
    float s = in_b[e];
    for (int c = 0; c < CC; ++c) s += pr[k * CC + c] * w[c];
    qkv[i] = s;
  }
  __syncthreads();

  const float scale = 0.25f;  // 1/sqrt(DH)
  for (int i = tid; i < HH * KK * KK; i += 256) {
    int h = i / (KK * KK), r = i % (KK * KK);
    int qi = r / KK, kj = r % KK;
    const float* qv = &qkv[qi * 3 * CC + h * DH];           // q slice
    const float* kv = &qkv[kj * 3 * CC + CC + h * DH];      // k slice
    float s = 0.0f;
#pragma unroll
    for (int d = 0; d < DH; ++d) s += qv[d] * kv[d];
    sc[i] = s * scale;
  }
  __syncthreads();

  if (tid < HH * KK) {  // softmax over last dim, one (h,qi) row per thread
    float* row = &sc[tid * KK];
    float mx = row[0];
    for (int j = 1; j < KK; ++j) mx = fmaxf(mx, row[j]);
    float sum = 0.0f;
    for (int j = 0; j < KK; ++j) { float e = __expf(row[j] - mx); row[j] = e; sum += e; }
    float inv = 1.0f / sum;
    for (int j = 0; j < KK; ++j) row[j] *= inv;
  }
  __syncthreads();

  for (int i = tid; i < KK * CC; i += 256) {  // o = attn @ v, merged heads
    int qi = i / CC, cc = i % CC;
    int h = cc >> 4, d = cc & 15;
    float s = 0.0f;
#pragma unroll
    for (int j = 0; j < KK; ++j)
      s += sc[(h * KK + qi) * KK + j] * qkv[j * 3 * CC + 2 * CC + h * DH + d];
    ob[i] = s;
  }
  __syncthreads();

  for (int i = tid; i < KK * CC; i += 256) {  // attended = o @ out_w^T + out_b
    int k = i / CC, co = i % CC;
    const float* w = out_w + (size_t)co * CC;
    float s = out_b[co];
    for (int c = 0; c < CC; ++c) s += ob[k * CC + c] * w[c];
    attended[(size_t)b * KK * CC + i] = s;
  }
}

// ---------------------------------------------------------------------------
// Kernel 5: out[b,c,p] = x[b,c,p] + attended[b, nearest[p], c]
// One (b,c) row per blockIdx.y; 14-entry LDS table; float4 streaming.
// ---------------------------------------------------------------------------
__global__ __launch_bounds__(256)
void scatter_kernel(const float* __restrict__ x, const float* __restrict__ attended,
                    const int* __restrict__ nearest, float* __restrict__ out) {
  int bc = blockIdx.y;                 // 0 .. BB*CC-1
  int b = bc / CC, c = bc % CC;
  __shared__ float tab[16];
  int tid = threadIdx.x;
  if (tid < KK) tab[tid] = attended[((size_t)b * KK + tid) * CC + c];
  __syncthreads();

  const float4* xr = (const float4*)(x + (size_t)bc * PP);
  const int4*   nr = (const int4*)nearest;
  float4*       orow = (float4*)(out + (size_t)bc * PP);
  const int P4 = PP / 4;               // PP % 4 == 0
  for (int i = blockIdx.x * 256 + tid; i < P4; i += gridDim.x * 256) {
    float4 xv = xr[i];
    int4   nv = nr[i];
    float4 o;
    o.x = xv.x + tab[nv.x];
    o.y = xv.y + tab[nv.y];
    o.z = xv.z + tab[nv.z];
    o.w = xv.w + tab[nv.w];
    orow[i] = o;
  }
}

// ---------------------------------------------------------------------------
// Launch
// ---------------------------------------------------------------------------
extern "C" void kernel_launch(void* const* d_in, const int* in_sizes, int n_in,
                              void* d_out, int out_size, void* d_ws, size_t ws_size,
                              hipStream_t stream) {
  const float* x        = (const float*)d_in[0];
  const float* points   = (const float*)d_in[1];
  const float* cavities = (const float*)d_in[2];
  const float* w1       = (const float*)d_in[3];
  const float* b1       = (const float*)d_in[4];
  const float* w2       = (const float*)d_in[5];
  const float* b2       = (const float*)d_in[6];
  const float* in_w     = (const float*)d_in[7];
  const float* in_b     = (const float*)d_in[8];
  const float* out_w    = (const float*)d_in[9];
  const float* out_b    = (const float*)d_in[10];
  float* out = (float*)d_out;

  // workspace layout (256B aligned slices)
  char* ws = (char*)d_ws;
  size_t off = 0;
  auto alloc = [&](size_t bytes) { char* p = ws + off; off += (bytes + 255) & ~(size_t)255; return p; };
  float*    sums     = (float*)   alloc((size_t)BB * KK * CC * 4);  // 57344
  float*    counts   = (float*)   alloc(16 * 4);
  float*    proc     = (float*)   alloc((size_t)BB * KK * CC * 4);
  float*    attended = (float*)   alloc((size_t)BB * KK * CC * 4);
  unsigned* maskbits = (unsigned*)alloc((size_t)PP * 4);
  int*      nearest  = (int*)     alloc((size_t)PP * 4);
  (void)ws_size; (void)in_sizes; (void)n_in; (void)out_size;

  // zero the atomic-accumulated buffers every call (deterministic, capturable)
  hipMemsetAsync(sums, 0, (size_t)BB * KK * CC * 4 + 256, stream);  // covers counts too

  prep_kernel<<<dim3((PP + 255) / 256), dim3(256), 0, stream>>>(
      points, cavities, maskbits, nearest, counts);

  pool_kernel<<<dim3(BB * (CC / 16), 16), dim3(256), 0, stream>>>(
      x, maskbits, sums);

  mlp_kernel<<<dim3(BB * KK), dim3(256), 0, stream>>>(
      sums, counts, w1, b1, w2, b2, proc);

  attn_kernel<<<dim3(BB), dim3(256), 0, stream>>>(
      proc, in_w, in_b, out_w, out_b, attended);

  scatter_kernel<<<dim3(8, BB * CC), dim3(256), 0, stream>>>(
      x, attended, nearest, out);
}